// MBWTConv2d_5617817224164
// MI455X (gfx1250) — compile-verified
//
#include <hip/hip_runtime.h>
#include <hip/hip_bf16.h>
#include <math.h>

typedef float v2f __attribute__((ext_vector_type(2)));
typedef float v8f __attribute__((ext_vector_type(8)));

static constexpr int Bsz = 4;
static constexpr int C   = 96;
static constexpr int Hh  = 128;
static constexpr int Ww  = 128;
static constexpr int L   = Hh * Ww;      // 16384
static constexpr int K   = 2;
static constexpr int R   = 6;
static constexpr int D8  = R + 2;        // R + 2*N, N=1
static constexpr int MT  = 3;            // M-tiles per wave in GEMM
static constexpr int G   = 2;            // channels per scan block

// ---------------------------------------------------------------------------
// WMMA GEMM: Y[b,o,l] (opt. * scale[o]) = sum_c W[o,c] * X[b,c,l]
// fp32 path: V_WMMA_F32_16X16X4_F32. Each wave owns a 48(M)x16(N) tile:
// one B-fragment load feeds MT=3 WMMAs (A-tiles staged in LDS), i.e. 72
// WMMA issues per wave against 48 activation loads -> 3x less L2 traffic.
// block = 128 threads (4 waves, 4 N-tiles). grid = (L/64, O/48, B).
// ---------------------------------------------------------------------------
__global__ __launch_bounds__(128)
void wmma_gemm_kernel(const float* __restrict__ Wm,   // O x 96 row-major
                      const float* __restrict__ X,    // B x 96 x L
                      float* __restrict__ Y,          // B x O x L
                      int O, const float* __restrict__ scale) {
  __shared__ float Ws[MT * 16 * 96];                  // 48x96 = 18 KB
  const int b     = blockIdx.z;
  const int m0    = blockIdx.y * (16 * MT);
  const int wave  = threadIdx.x >> 5;
  const int lane  = threadIdx.x & 31;

  // cooperative stage of the 48x96 weight tile
  for (int i = threadIdx.x; i < MT * 16 * 96; i += 128)
    Ws[i] = Wm[(m0 + (i / 96)) * 96 + (i % 96)];
  __syncthreads();

  const int n0   = (blockIdx.x * 4 + wave) * 16;
  const int mrow = lane & 15;                 // A-frag row (within 16-row tile)
  const int kb   = (lane >> 4) * 2;           // A/B-frag K sub-offset
  const int ncol = n0 + (lane & 15);          // B/D-frag column
  const float* __restrict__ Xb = X + (size_t)b * 96 * L;

  v8f acc[MT] = {};
#pragma unroll
  for (int k0 = 0; k0 < 96; k0 += 4) {
    v2f bb;
    bb.x = Xb[(size_t)(k0 + kb)     * L + ncol];
    bb.y = Xb[(size_t)(k0 + kb + 1) * L + ncol];
#pragma unroll
    for (int mt = 0; mt < MT; ++mt) {
      v2f a;
      a.x = Ws[(mt * 16 + mrow) * 96 + k0 + kb];
      a.y = Ws[(mt * 16 + mrow) * 96 + k0 + kb + 1];
      acc[mt] = __builtin_amdgcn_wmma_f32_16x16x4_f32(false, a, false, bb,
                                                      (short)0, acc[mt],
                                                      false, false);
    }
  }

  const int mhalf = (lane >> 4) * 8;          // D layout: VGPR r -> row r (+8 hi lanes)
#pragma unroll
  for (int mt = 0; mt < MT; ++mt) {
    const int mb = m0 + mt * 16 + mhalf;
    float* __restrict__ Yb = Y + ((size_t)b * O + mb) * L + ncol;
#pragma unroll
    for (int r = 0; r < 8; ++r) {
      float v = acc[mt][r];
      if (scale) v *= scale[mb + r];
      Yb[(size_t)r * L] = v;
    }
  }
}

// ---------------------------------------------------------------------------
// Depthwise 3x3 SAME conv + bias + SiLU on xz[:, :96] -> u (B,96,L)
// ---------------------------------------------------------------------------
__global__ __launch_bounds__(256)
void conv_silu_kernel(const float* __restrict__ xz,
                      const float* __restrict__ cw,  // (96,1,3,3)
                      const float* __restrict__ cb,  // (96)
                      float* __restrict__ u) {
  int idx = blockIdx.x * 256 + threadIdx.x;          // B*96*L
  if (idx >= Bsz * C * L) return;
  int w = idx % Ww;
  int h = (idx / Ww) % Hh;
  int c = (idx / L) % C;
  int b = idx / (C * L);
  const float* __restrict__ src = xz + ((size_t)b * (2 * C) + c) * L;
  const float* __restrict__ kc  = cw + c * 9;
  float acc = cb[c];
#pragma unroll
  for (int dy = -1; dy <= 1; ++dy) {
    int hy = h + dy;
    if (hy < 0 || hy >= Hh) continue;
#pragma unroll
    for (int dx = -1; dx <= 1; ++dx) {
      int wx = w + dx;
      if (wx < 0 || wx >= Ww) continue;
      acc += kc[(dy + 1) * 3 + (dx + 1)] * src[hy * Ww + wx];
    }
  }
  u[idx] = acc / (1.f + expf(-acc));                 // SiLU
}

// ---------------------------------------------------------------------------
// x_proj: dtBC[b,k,d,pos] = sum_c u[b,c,lmap(pos)] * x_proj_w[k,d,c]
// pos is the scan coordinate (direction k=1 reads u reversed).
// ---------------------------------------------------------------------------
__global__ __launch_bounds__(256)
void xproj_kernel(const float* __restrict__ u,
                  const float* __restrict__ xpw,     // (K, 8, 96)
                  float* __restrict__ dtBC) {        // (B, K, 8, L)
  int idx = blockIdx.x * 256 + threadIdx.x;          // B*K*L
  if (idx >= Bsz * K * L) return;
  int pos = idx % L;
  int k   = (idx / L) % K;
  int b   = idx / (K * L);
  int l   = k ? (L - 1 - pos) : pos;
  const float* __restrict__ wb = xpw + k * D8 * 96;
  const float* __restrict__ ub = u + (size_t)b * C * L + l;
  float acc[D8] = {0.f, 0.f, 0.f, 0.f, 0.f, 0.f, 0.f, 0.f};
  for (int c = 0; c < 96; ++c) {
    float uv = ub[(size_t)c * L];
#pragma unroll
    for (int d = 0; d < D8; ++d) acc[d] += wb[d * 96 + c] * uv;
  }
  float* __restrict__ ob = dtBC + ((size_t)(b * K + k)) * D8 * L + pos;
#pragma unroll
  for (int d = 0; d < D8; ++d) ob[(size_t)d * L] = acc[d];
}

// ---------------------------------------------------------------------------
// Chunked parallel linear-recurrence scan. N=1 => scalar recurrence.
// One block per (b,k,channel-pair); 256 threads x 64-element chunks.
// G=2 channels share the 8 dtBC row loads (they are channel-invariant),
// halving L2 traffic. Pass1: compose chunk into affine (a,b); LDS
// Hillis-Steele scan over affine maps; Pass2: replay with incoming prefix.
// delta/softplus/dA/dBu computed inline from dtBC (4 MB, L2-resident).
// ---------------------------------------------------------------------------
__global__ __launch_bounds__(256)
void scan_kernel(const float* __restrict__ u,
                 const float* __restrict__ dtBC,
                 const float* __restrict__ dtw,     // (K, 96, 6)
                 const float* __restrict__ dtb,     // (K, 96)
                 const float* __restrict__ A_logs,  // (K*96, 1)
                 const float* __restrict__ Ds,      // (K*96)
                 float* __restrict__ ys) {          // (B, K, 96, L) pos-indexed
  __shared__ float sA[G * 256];
  __shared__ float sB[G * 256];
  const int CG  = C / G;                     // 48 channel-groups
  const int bkg = blockIdx.x;                // b*K*CG + k*CG + grp
  const int c0  = (bkg % CG) * G;
  const int k   = (bkg / CG) % K;
  const int b   = bkg / (K * CG);
  const int t   = threadIdx.x;
  const int CH  = L / 256;                   // 64

  float wr[G][R], bias[G], Aval[G], Dval[G];
#pragma unroll
  for (int g = 0; g < G; ++g) {
    const int c = c0 + g;
#pragma unroll
    for (int r = 0; r < R; ++r) wr[g][r] = dtw[(k * C + c) * R + r];
    bias[g] = dtb[k * C + c];
    Aval[g] = -expf(A_logs[k * C + c]);
    Dval[g] = Ds[k * C + c];
  }
  const float* __restrict__ dts = dtBC + ((size_t)(b * K + k)) * D8 * L;
  const float* __restrict__ ub  = u + ((size_t)b * C + c0) * L;

  auto elem = [&](int pos, float* dA, float* dBu, float& Csv, float* uval) {
    const int l = k ? (L - 1 - pos) : pos;
    float dv[R];
#pragma unroll
    for (int r = 0; r < R; ++r) dv[r] = dts[(size_t)r * L + pos];
    const float Bsv = dts[(size_t)R * L + pos];
    Csv             = dts[(size_t)(R + 1) * L + pos];
#pragma unroll
    for (int g = 0; g < G; ++g) {
      uval[g] = ub[(size_t)g * L + l];
      float x = bias[g];
#pragma unroll
      for (int r = 0; r < R; ++r) x += wr[g][r] * dv[r];
      float delta = (x > 20.f) ? x : log1pf(expf(x));
      dA[g]  = expf(delta * Aval[g]);
      dBu[g] = delta * uval[g] * Bsv;
    }
  };

  // pass 1: compose chunk -> (aC, bC) with h_out = aC*h_in + bC
  float aC[G], bC[G];
#pragma unroll
  for (int g = 0; g < G; ++g) { aC[g] = 1.f; bC[g] = 0.f; }
  const int base = t * CH;
  for (int i = 0; i < CH; ++i) {
    float dA[G], dBu[G], Csv, uv[G];
    elem(base + i, dA, dBu, Csv, uv);
#pragma unroll
    for (int g = 0; g < G; ++g) {
      aC[g] = dA[g] * aC[g];
      bC[g] = dA[g] * bC[g] + dBu[g];
    }
  }
#pragma unroll
  for (int g = 0; g < G; ++g) { sA[g * 256 + t] = aC[g]; sB[g * 256 + t] = bC[g]; }
  __syncthreads();

  // inclusive scan over affine maps (composition: later ∘ earlier)
  for (int off = 1; off < 256; off <<= 1) {
    float pa[G], pb[G], ca[G], cb[G];
#pragma unroll
    for (int g = 0; g < G; ++g) {
      pa[g] = (t >= off) ? sA[g * 256 + t - off] : 1.f;
      pb[g] = (t >= off) ? sB[g * 256 + t - off] : 0.f;
      ca[g] = sA[g * 256 + t];
      cb[g] = sB[g * 256 + t];
    }
    __syncthreads();
#pragma unroll
    for (int g = 0; g < G; ++g) {
      sA[g * 256 + t] = ca[g] * pa[g];
      sB[g * 256 + t] = ca[g] * pb[g] + cb[g];
    }
    __syncthreads();
  }
  float h[G];
#pragma unroll
  for (int g = 0; g < G; ++g)
    h[g] = (t == 0) ? 0.f : sB[g * 256 + t - 1];   // exclusive prefix on h0=0

  // pass 2: replay with prefix; fuse y = Cs*h + Ds*xs
  float* __restrict__ yb = ys + ((size_t)((b * K + k) * C + c0)) * L;
  for (int i = 0; i < CH; ++i) {
    float dA[G], dBu[G], Csv, uv[G];
    elem(base + i, dA, dBu, Csv, uv);
#pragma unroll
    for (int g = 0; g < G; ++g) {
      h[g] = dA[g] * h[g] + dBu[g];
      yb[(size_t)g * L + base + i] = Csv * h[g] + Dval[g] * uv[g];
    }
  }
}

// ---------------------------------------------------------------------------
// Direction merge + channel LayerNorm + SiLU(z) gate -> g (B,96,L)
// ---------------------------------------------------------------------------
__global__ __launch_bounds__(256)
void norm_gate_kernel(const float* __restrict__ ys,
                      const float* __restrict__ xz,
                      const float* __restrict__ nw,
                      const float* __restrict__ nb,
                      float* __restrict__ g) {
  int idx = blockIdx.x * 256 + threadIdx.x;          // B*L
  if (idx >= Bsz * L) return;
  int l = idx % L;
  int b = idx / L;
  int lr = L - 1 - l;
  const float* __restrict__ y0 = ys + (size_t)(b * K + 0) * C * L;
  const float* __restrict__ y1 = ys + (size_t)(b * K + 1) * C * L;
  float s = 0.f, s2 = 0.f;
  for (int c = 0; c < C; ++c) {
    float v = y0[(size_t)c * L + l] + y1[(size_t)c * L + lr];
    s += v; s2 += v * v;
  }
  float mu  = s * (1.f / C);
  float var = s2 * (1.f / C) - mu * mu;
  float inv = rsqrtf(var + 1e-5f);
  const float* __restrict__ zb = xz + ((size_t)b * (2 * C) + C) * L;
  float* __restrict__ gb = g + (size_t)b * C * L;
  for (int c = 0; c < C; ++c) {
    float v  = y0[(size_t)c * L + l] + y1[(size_t)c * L + lr];
    v = (v - mu) * inv * nw[c] + nb[c];
    float zv = zb[(size_t)c * L + l];
    gb[(size_t)c * L + l] = v * (zv / (1.f + expf(-zv)));
  }
}

// ---------------------------------------------------------------------------
extern "C" void kernel_launch(void* const* d_in, const int* in_sizes, int n_in,
                              void* d_out, int out_size, void* d_ws, size_t ws_size,
                              hipStream_t stream) {
  const float* x         = (const float*)d_in[0];   // (4,96,128,128)
  const float* in_proj_w = (const float*)d_in[1];   // (192,96)
  const float* conv_w    = (const float*)d_in[2];   // (96,1,3,3)
  const float* conv_b    = (const float*)d_in[3];   // (96)
  const float* x_proj_w  = (const float*)d_in[4];   // (2,8,96)
  const float* dt_proj_w = (const float*)d_in[5];   // (2,96,6)
  const float* dt_proj_b = (const float*)d_in[6];   // (2,96)
  const float* A_logs    = (const float*)d_in[7];   // (192,1)
  const float* Ds        = (const float*)d_in[8];   // (192)
  const float* out_nw    = (const float*)d_in[9];   // (96)
  const float* out_nb    = (const float*)d_in[10];  // (96)
  const float* out_proj_w= (const float*)d_in[11];  // (96,96)
  const float* scale_w   = (const float*)d_in[12];  // (1,96,1,1)
  float* out = (float*)d_out;

  float* ws = (float*)d_ws;
  float* xz   = ws;                                     // B*192*L
  float* u    = xz   + (size_t)Bsz * 2 * C * L;         // B*96*L
  float* dtBC = u    + (size_t)Bsz * C * L;             // B*K*8*L
  float* ys   = dtBC + (size_t)Bsz * K * D8 * L;        // B*K*96*L
  float* g    = ys   + (size_t)Bsz * K * C * L;         // B*96*L

  // 1) in_proj: xz = in_proj_w @ x    (WMMA f32, 48-row M blocks)
  wmma_gemm_kernel<<<dim3(L / 64, (2 * C) / (16 * MT), Bsz), 128, 0, stream>>>(
      in_proj_w, x, xz, 2 * C, nullptr);

  // 2) depthwise conv + SiLU -> u
  conv_silu_kernel<<<(Bsz * C * L) / 256, 256, 0, stream>>>(xz, conv_w, conv_b, u);

  // 3) x_proj -> dtBC (pos-indexed per direction)
  xproj_kernel<<<(Bsz * K * L) / 256, 256, 0, stream>>>(u, x_proj_w, dtBC);

  // 4) chunked parallel selective scan -> ys (channel pairs share dtBC)
  scan_kernel<<<Bsz * K * (C / G), 256, 0, stream>>>(u, dtBC, dt_proj_w,
                                                     dt_proj_b, A_logs, Ds, ys);

  // 5) merge directions + layernorm + gate -> g
  norm_gate_kernel<<<(Bsz * L) / 256, 256, 0, stream>>>(ys, xz, out_nw, out_nb, g);

  // 6) out_proj (+ per-channel scale) -> out   (WMMA f32, 48-row M blocks)
  wmma_gemm_kernel<<<dim3(L / 64, C / (16 * MT), Bsz), 128, 0, stream>>>(
      out_proj_w, g, out, C, scale_w);
}